// WaveDec_15685220565767
// MI455X (gfx1250) — compile-verified
//
#include <hip/hip_runtime.h>
#include <stdint.h>

// ---------------------------------------------------------------------------
// 11-level DB4 DWT, fully fused per row.
// Traffic = 1 read + 1 write of 64MB (~5.5us @ 23.3 TB/s roofline). Compute
// via v_wmma_f32_16x16x4_f32; row load via global_load_async_to_lds_b128.
// B-operand fetched as one ds_load_b64 and software-pipelined so the WMMA
// does not stall on a dscnt==0 wait each tile.
// ---------------------------------------------------------------------------

typedef __attribute__((ext_vector_type(2))) float v2f;
typedef __attribute__((ext_vector_type(8))) float v8f;
typedef __attribute__((ext_vector_type(4))) int   v4i;

#define AS1 __attribute__((address_space(1)))
#define AS3 __attribute__((address_space(3)))

__device__ __forceinline__ void async_load_b128(const float* gsrc, float* ldst) {
#if __has_builtin(__builtin_amdgcn_global_load_async_to_lds_b128)
  __builtin_amdgcn_global_load_async_to_lds_b128(
      (AS1 v4i*)gsrc, (AS3 v4i*)ldst, 0, 0);
#else
  unsigned loff = (unsigned)(uintptr_t)(AS3 void*)ldst;
  asm volatile("global_load_async_to_lds_b128 %0, %1, off"
               :: "v"(loff), "v"(gsrc)
               : "memory");
#endif
}

__device__ __forceinline__ void wait_async0() {
#if __has_builtin(__builtin_amdgcn_s_wait_asynccnt)
  __builtin_amdgcn_s_wait_asynccnt(0);
#else
  asm volatile("s_wait_asynccnt 0" ::: "memory");
#endif
}

__global__ __launch_bounds__(256)
void dwt_db4_fused(const float* __restrict__ x, float* __restrict__ out) {
  constexpr int N0 = 8192;
  constexpr int LEVELS = 11;
  const float C0 = 0.4829629131445341f;
  const float C1 = 0.8365163037378079f;
  const float C2 = 0.2241438680420134f;
  const float C3 = -0.1294095225512604f;

  __shared__ float lds0[N0];       // even levels read here
  __shared__ float lds1[N0 / 2];   // odd levels read here

  const int tid  = threadIdx.x;
  const int lane = tid & 31;       // wave32
  const int wave = tid >> 5;       // 8 waves / block
  const float* gx   = x   + (size_t)blockIdx.x * N0;
  float*       gout = out + (size_t)blockIdx.x * N0;

  // ---- Stage 1: async DMA the whole row into LDS (ASYNCcnt path) ----------
  for (int c = tid; c < N0 / 4; c += 256)
    async_load_b128(gx + 4 * c, &lds0[4 * c]);
  wait_async0();
  __syncthreads();

  // ---- A matrix (16x4 f32): rows 0/1 = DB4 lowpass / highpass filters -----
  // 32-bit A layout: lane L<16 holds A[L,0..1] in (x,y); lane L+16 holds A[L,2..3].
  v2f A = {0.f, 0.f};
  if      (lane == 0)  { A.x = C0; A.y = C1;  }   // A[0,0..1]
  else if (lane == 16) { A.x = C2; A.y = C3;  }   // A[0,2..3]
  else if (lane == 1)  { A.x = C3; A.y = -C2; }   // A[1,0..1]
  else if (lane == 17) { A.x = C1; A.y = -C0; }   // A[1,2..3]

  const int j  = lane & 15;        // output column within the 16-wide tile
  const int kh = lane >> 4;        // 0 for lanes 0-15 (K=0,1), 1 for 16-31 (K=2,3)

  for (int lev = 0; lev < LEVELS; ++lev) {
    const int half = (N0 >> lev) >> 1;
    const int hm   = half - 1;          // pair-index wrap mask (power of two)
    const v2f* src2 = (const v2f*)((lev & 1) ? lds1 : lds0);  // pairs (x[2p],x[2p+1])
    float*     dst  = (lev & 1) ? lds0 : lds1;

    // Each wave computes 16 (a,d) pairs per WMMA. k0 is wave-uniform, so the
    // guard and loop keep EXEC all-ones at every wmma. B column j covers the
    // window x[2k .. 2k+3] (mod n): lanes 0-15 load pair k, lanes 16-31 pair k+1.
    int k0 = wave * 16;
    if (k0 < half) {
      v2f B = src2[(k0 + j + kh) & hm];
      for (; k0 < half; k0 += 8 * 16) {
        // Prefetch next tile's B (mod-masked, safe overshoot on last iter).
        v2f Bn = src2[(k0 + 128 + j + kh) & hm];
        v8f Cacc = {0.f, 0.f, 0.f, 0.f, 0.f, 0.f, 0.f, 0.f};
        v8f D = __builtin_amdgcn_wmma_f32_16x16x4_f32(
            false, A, false, B, (short)0, Cacc, false, false);
        // D[0] lanes 0-15 = row M=0 = a_k ; D[1] lanes 0-15 = row M=1 = d_k.
        const int k = k0 + j;
        if (lane < 16 && k < half) {
          dst[k]         = D[0];   // approximation -> ping-pong LDS
          gout[half + k] = D[1];   // detail coeffs are final: stream to output
        }
        B = Bn;
      }
    }
    __syncthreads();
  }

  // Final approximation (length N0 >> LEVELS = 4) sits in lds1 (last dst).
  const int nfin = N0 >> LEVELS;
  if (tid < nfin) gout[tid] = lds1[tid];
}

extern "C" void kernel_launch(void* const* d_in, const int* in_sizes, int n_in,
                              void* d_out, int out_size, void* d_ws, size_t ws_size,
                              hipStream_t stream) {
  (void)n_in; (void)d_ws; (void)ws_size; (void)out_size;
  const float* x = (const float*)d_in[0];   // (BATCH, 8192) f32
  float* out = (float*)d_out;               // (BATCH, 8192) f32
  const int batch = in_sizes[0] / 8192;
  dwt_db4_fused<<<batch, 256, 0, stream>>>(x, out);
}